// RoIPooling_28063316312494
// MI455X (gfx1250) — compile-verified
//
#include <hip/hip_runtime.h>
#include <hip/hip_bf16.h>
#include <stdint.h>

// Problem constants (from reference)
#define B_   16
#define H_   64
#define W_   64
#define C_   512
#define R_   256
#define NR_  64
#define PH_  7
#define PW_  7
#define IOU_THR_ 0.4f

#define POOL_T 128           // 4 wave32s; each lane owns 4 channels (float4)
#define PIPE_D 8             // async pipeline depth (pixels in flight per wave)

#define AS1 __attribute__((address_space(1)))
#define AS3 __attribute__((address_space(3)))

// The async-copy builtin takes pointers to 16-byte int vectors (v4i) in
// global (AS1) / LDS (AS3) address spaces.
typedef int v4i_ __attribute__((vector_size(16)));

#if defined(__HIP_DEVICE_COMPILE__)
  #if __has_builtin(__builtin_amdgcn_global_load_async_to_lds_b128)
    #define ASYNC_COPY16(gp, lp)                                              \
      __builtin_amdgcn_global_load_async_to_lds_b128(                         \
          (AS1 v4i_*)(gp), (AS3 v4i_*)(lp), 0, 0)
  #else
    #define ASYNC_COPY16(gp, lp)                                              \
      asm volatile("global_load_async_to_lds_b128 %0, %1, off"                \
                   :: "v"((unsigned)(unsigned long long)(AS3 v4i_*)(lp)),     \
                      "v"((const void*)(gp))                                  \
                   : "memory")
  #endif
  #if __has_builtin(__builtin_amdgcn_s_wait_asynccnt)
    #define WAIT_ASYNC(n) do { __builtin_amdgcn_s_wait_asynccnt(n); \
                               asm volatile("" ::: "memory"); } while (0)
  #else
    #define WAIT_ASYNC(n) asm volatile("s_wait_asynccnt %0" :: "i"(n) : "memory")
  #endif
#else
  #define ASYNC_COPY16(gp, lp) ((void)0)
  #define WAIT_ASYNC(n)        ((void)0)
#endif

__device__ __forceinline__ float4 vmax4(float4 a, float4 b) {
  float4 r;
  r.x = fmaxf(a.x, b.x); r.y = fmaxf(a.y, b.y);
  r.z = fmaxf(a.z, b.z); r.w = fmaxf(a.w, b.w);
  return r;
}

// _fix_axis with Python floor-division semantics (pad may be negative; >>1 floors)
__device__ __forceinline__ void fix_axis(int* mn_, int* mx_, int ps, int fs) {
  int mn = *mn_, mx = *mx_;
  int pad = ps - (mx - mn);
  bool fix_min = mn < (pad >> 1);
  bool fix_max = (fs - mx) < ((1 + pad) >> 1);
  bool sym = (pad > 0) && !(fix_min || fix_max);
  int omin = sym ? (mn - (pad >> 1)) : mn;
  int omax = sym ? (mx + ((1 + pad) >> 1)) : mx;
  if ((pad > 0) && fix_min) { omin = 0;       omax = ps; }
  if ((pad > 0) && fix_max) { omin = fs - ps; omax = fs; }
  *mn_ = omin; *mx_ = omax;
}

// ---------------------------------------------------------------------------
// Kernel 1: per-batch greedy NMS + ROI clipping.
// One block per batch; lane t owns ROI t. The greedy scan is inherently
// serial in i (keep[i] depends on all keep[<i]); per step, all 256 lanes test
// IoU(i, t) in parallel and OR into an LDS flag (ds atomic).
// ---------------------------------------------------------------------------
__global__ __launch_bounds__(R_) void nms_clip_kernel(
    const float* __restrict__ roi,   // [B, R, 4] (x, y, w, h)
    float* __restrict__ roi_out,     // [B, NR, 4] (float-encoded ints), tail of d_out
    int* __restrict__ roi_ws) {      // [B, NR, 4] ints for kernel 2
  const int b = blockIdx.x, t = threadIdx.x;
  __shared__ float sx1[R_], sy1[R_], sx2[R_], sy2[R_], sar[R_];
  __shared__ int   skeep[R_];
  __shared__ int   sidx[NR_];
  __shared__ int   sflag;

  const float* rp = roi + ((size_t)b * R_ + t) * 4;
  float x = rp[0], y = rp[1], w = rp[2], h = rp[3];
  sx1[t] = x; sy1[t] = y; sx2[t] = x + w; sy2[t] = y + h; sar[t] = w * h;
  skeep[t] = 0;
  __syncthreads();

  for (int i = 0; i < R_; ++i) {
    if (t == 0) sflag = 0;
    __syncthreads();
    if (skeep[t]) {  // only already-kept ROIs can suppress row i
      float ih = fmaxf(fminf(sy2[i], sy2[t]) - fmaxf(sy1[i], sy1[t]), 0.0f);
      float iw = fmaxf(fminf(sx2[i], sx2[t]) - fmaxf(sx1[i], sx1[t]), 0.0f);
      float inter = ih * iw;
      float uni = sar[i] + sar[t] - inter;
      float iou = (uni > 0.0f) ? (inter / uni) : 0.0f;
      if (iou > IOU_THR_) atomicOr(&sflag, 1);
    }
    __syncthreads();
    if (t == i) skeep[t] = (sflag == 0) ? 1 : 0;
    __syncthreads();
  }

  // rank/compact + fill (fill[j] = R - NR + j)
  if (t == 0) {
    for (int k = 0; k < NR_; ++k) sidx[k] = (R_ - NR_) + k;
    int rank = 0;
    for (int k = 0; k < R_; ++k)
      if (skeep[k]) { if (rank < NR_) sidx[rank] = k; ++rank; }
  }
  __syncthreads();

  if (t < NR_) {
    int s = sidx[t];
    int xmin = (int)fmaxf(0.0f, sx1[s]);
    int ymin = (int)fmaxf(0.0f, sy1[s]);
    int xmax = (int)fminf((float)W_, sx2[s]);
    int ymax = (int)fminf((float)H_, sy2[s]);
    fix_axis(&xmin, &xmax, PW_, W_);
    fix_axis(&ymin, &ymax, PH_, H_);
    const int o = (b * NR_ + t) * 4;
    roi_ws[o + 0] = xmin;          roi_ws[o + 1] = ymin;
    roi_ws[o + 2] = xmax - xmin;   roi_ws[o + 3] = ymax - ymin;
    roi_out[o + 0] = (float)xmin;          roi_out[o + 1] = (float)ymin;
    roi_out[o + 2] = (float)(xmax - xmin); roi_out[o + 3] = (float)(ymax - ymin);
  }
}

// ---------------------------------------------------------------------------
// Kernel 2: RoI pooling. One block per (b, roi). Lane t owns channels
// [4t, 4t+4). Pixels of the (h x w) ROI stream through an 8-slot LDS ring via
// the CDNA5 async copy engine; each lane consumes only the 16B it requested,
// so ASYNCcnt (wave-private, in-order for global loads) is the only sync.
// ---------------------------------------------------------------------------
__global__ __launch_bounds__(POOL_T) void roi_pool_kernel(
    const float* __restrict__ features,  // [B, H, W, C]
    const int*   __restrict__ roi_ws,    // [B*NR, 4]
    float*       __restrict__ pooled) {  // [B, NR, 7, 7, C]
  __shared__ float buf[PIPE_D * POOL_T * 4];  // 16 KB ring: slot = one pixel (2KB)

  const int br = blockIdx.x;
  const int t  = threadIdx.x;
  const int x = roi_ws[br * 4 + 0], y = roi_ws[br * 4 + 1];
  const int w = roi_ws[br * 4 + 2], h = roi_ws[br * 4 + 3];  // clipped => w,h >= 7
  const int b = br >> 6;

  const float* fmt  = features + (size_t)b * ((size_t)H_ * W_ * C_) + 4 * t;
  float*       outp = pooled   + (size_t)br * (PH_ * PW_ * C_) + 4 * t;
  float*       lb   = &buf[4 * t];

  const float NEGF = __int_as_float(0xff800000u);  // -inf
  const float4 neg4 = make_float4(NEGF, NEGF, NEGF, NEGF);
  float4 colmax = neg4, corner = neg4;
  float4 rs0 = neg4, rs1 = neg4, rs2 = neg4, rs3 = neg4, rs4 = neg4, rs5 = neg4;

  const int P = w * h;

  // ---- prologue: fill the pipeline ----
  int iri = 0, icj = 0, issued = 0;
  const int pre = (P < PIPE_D) ? P : PIPE_D;
  for (; issued < pre; ++issued) {
    const float* g = fmt + ((size_t)((y + iri) * W_ + (x + icj))) * C_;
    ASYNC_COPY16(g, lb + (issued & (PIPE_D - 1)) * (POOL_T * 4));
    if (++icj == w) { icj = 0; ++iri; }
  }

  int cri = 0, ccj = 0;
  auto consume = [&](int slot) {
    float4 v = *(const float4*)(lb + slot * (POOL_T * 4));  // ds_load_b128
    const bool last = (ccj == w - 1);                       // w>=7 => last col >= 6
    if (cri < 6) {
      if (ccj < 6) {
        *(float4*)(outp + (cri * PW_ + ccj) * C_) = v;      // block copy cell
      } else {
        colmax = vmax4(colmax, v);
      }
      if (last) { *(float4*)(outp + (cri * PW_ + 6) * C_) = colmax; colmax = neg4; }
    } else {
      if (ccj >= 6) {
        colmax = vmax4(colmax, v);
      } else if (ccj == 0) { rs0 = vmax4(rs0, v);
      } else if (ccj == 1) { rs1 = vmax4(rs1, v);
      } else if (ccj == 2) { rs2 = vmax4(rs2, v);
      } else if (ccj == 3) { rs3 = vmax4(rs3, v);
      } else if (ccj == 4) { rs4 = vmax4(rs4, v);
      } else               { rs5 = vmax4(rs5, v); }
      if (last) { corner = vmax4(corner, colmax); colmax = neg4; }
    }
    if (++ccj == w) { ccj = 0; ++cri; }
  };

  // ---- steady state: keep PIPE_D pixels in flight ----
  int p = 0;
  for (; issued < P; ++p) {
    WAIT_ASYNC(PIPE_D - 1);           // oldest in-flight pixel has landed in LDS
    consume(p & (PIPE_D - 1));
    const float* g = fmt + ((size_t)((y + iri) * W_ + (x + icj))) * C_;
    ASYNC_COPY16(g, lb + (issued & (PIPE_D - 1)) * (POOL_T * 4));
    ++issued;
    if (++icj == w) { icj = 0; ++iri; }
  }

  // ---- drain ----
  WAIT_ASYNC(0);
  for (; p < P; ++p) consume(p & (PIPE_D - 1));

  // ---- bottom row + corner ----
  *(float4*)(outp + (6 * PW_ + 0) * C_) = rs0;
  *(float4*)(outp + (6 * PW_ + 1) * C_) = rs1;
  *(float4*)(outp + (6 * PW_ + 2) * C_) = rs2;
  *(float4*)(outp + (6 * PW_ + 3) * C_) = rs3;
  *(float4*)(outp + (6 * PW_ + 4) * C_) = rs4;
  *(float4*)(outp + (6 * PW_ + 5) * C_) = rs5;
  *(float4*)(outp + (6 * PW_ + 6) * C_) = corner;
}

extern "C" void kernel_launch(void* const* d_in, const int* in_sizes, int n_in,
                              void* d_out, int out_size, void* d_ws, size_t ws_size,
                              hipStream_t stream) {
  const float* features = (const float*)d_in[0];   // [16,64,64,512] f32
  const float* roi      = (const float*)d_in[1];   // [16,256,4] f32
  float* out = (float*)d_out;
  // output = pooled [16,64,7,7,512] (25,690,112 floats) ++ roi_clipped [16,64,4]
  float* roi_out = out + (size_t)B_ * NR_ * PH_ * PW_ * C_;
  int* ws = (int*)d_ws;                             // 16 KB of clipped int ROIs

  nms_clip_kernel<<<B_, R_, 0, stream>>>(roi, roi_out, ws);
  roi_pool_kernel<<<B_ * NR_, POOL_T, 0, stream>>>(features, ws, out);
}